// MovingMaxMinVertical_87617332838552
// MI455X (gfx1250) — compile-verified
//
#include <hip/hip_runtime.h>

// Moving max (window 25) then moving min (window 25) along axis -2 of
// (8, 4096, 2048) f32, edge-replicated padding. Memory-bound: use the CDNA5
// async global->LDS engine to stream row segments while the VALU runs a
// van Herk / Gil-Werman two-stage scan (O(1) amortized compares per element).
// Outputs are stored non-temporally so the 192MB L2 keeps input halo rows
// for the neighboring row-chunk blocks.

#define LF    25                 // filter length
#define CH    152                // output rows per block (CH + 2*(LF-1) = 200 = 8*25)
#define M1    (CH + LF - 1)      // 176 stage-1 (max) outputs needed
#define NSEG  8                  // input segments of LF rows: NSEG*LF = 200
#define TC    128                // columns per block == blockDim.x
#define NROW  4096
#define NCOL  2048
#define NBATCH 8

__global__ __launch_bounds__(TC) void
moving_maxmin_kernel(const float* __restrict__ x, float* __restrict__ out)
{
    __shared__ float smem[2 * LF * TC];   // double-buffered 25x128 f32 segments

    const int tid     = threadIdx.x;
    const int colbase = blockIdx.x * TC;
    const int c       = colbase + tid;
    const int row0    = blockIdx.y * CH;
    const float* xb   = x   + (size_t)blockIdx.z * NROW * NCOL;
    float*       ob   = out + (size_t)blockIdx.z * NROW * NCOL;

    // Low 32 bits of a generic pointer to LDS == LDS byte offset on gfx1250.
    const unsigned lds0 = (unsigned)(size_t)(&smem[0]);

    // Async-load segment s (25 rows x 128 cols) into buffer (s&1).
    // Wave w loads rows {w, w+4, ...}; lane loads a 16-byte (4-col) strip.
    const int w    = tid >> 5;
    const int lane = tid & 31;
    auto issue_seg = [&](int s) {
        const unsigned lbase = lds0 + (unsigned)((s & 1) * (LF * TC * 4));
#pragma unroll
        for (int rr = 0; rr < LF; rr += 4) {
            const int r = rr + w;            // wave-uniform predicate below
            if (r < LF) {
                int g = row0 + s * LF + r - (LF - 1);
                g = g < 0 ? 0 : (g > NROW - 1 ? NROW - 1 : g);
                const unsigned voff  = (unsigned)((g * NCOL + colbase + lane * 4) * 4);
                const unsigned laddr = lbase + (unsigned)(r * TC * 4 + lane * 16);
                asm volatile("global_load_async_to_lds_b128 %0, %1, %2"
                             :: "v"(laddr), "v"(voff), "s"(xb)
                             : "memory");
            }
        }
    };

    // ---- two-stage streaming van Herk automaton (registers only) ----
    float B1[LF], S1[LF];        // stage-1 (max): segment buffer + suffix-max
    float B2[LF], S2[LF];        // stage-2 (min): segment buffer + suffix-min
    float p2 = __builtin_inff(); // stage-2 running prefix-min of current segment

    // Consume one stage-1 output (mx stream element u2); emits one final
    // output when u2 >= LF. u2 is compile-time constant at every call site.
    auto consume = [&](int u2, float v) {
        const int t2 = u2 % LF;
        if (u2 >= LF) {
            const int q2  = u2 - LF;            // 0..CH-1
            const float o = fminf(S2[t2], p2);
            const int row = row0 + q2;
            if (row < NROW)                     // block-uniform predicate
                __builtin_nontemporal_store(o, &ob[(size_t)row * NCOL + c]);
        }
        p2 = fminf(p2, v);
        B2[t2] = v;
        if (t2 == LF - 1) {                     // rebuild suffix-min, new segment
            S2[LF - 1] = B2[LF - 1];
#pragma unroll
            for (int tt = LF - 2; tt >= 0; --tt) S2[tt] = fminf(B2[tt], S2[tt + 1]);
            p2 = __builtin_inff();
        }
    };

    // ---- pipeline prologue: segments 0 and 1 in flight ----
    issue_seg(0);
    issue_seg(1);

    // Wave0 issues 7 ops/segment, others 6; wait<=6 guarantees oldest segment done.
    asm volatile("s_wait_asynccnt %0" :: "n"(6) : "memory");
    __syncthreads();                        // seg0 visible to all waves
#pragma unroll
    for (int t = 0; t < LF; ++t) B1[t] = smem[t * TC + tid];
    S1[LF - 1] = B1[LF - 1];
#pragma unroll
    for (int t = LF - 2; t >= 0; --t) S1[t] = fmaxf(B1[t], S1[t + 1]);
    __syncthreads();                        // all waves done reading buf0
    issue_seg(2);                           // -> buf0

    // ---- main loop: read segment j while emitting segment j-1 outputs ----
#pragma unroll
    for (int j = 1; j <= NSEG; ++j) {
        const bool has_read = (j < NSEG);
        if (has_read) {
            if (j < NSEG - 1) {
                asm volatile("s_wait_asynccnt %0" :: "n"(6) : "memory");
            } else {
                asm volatile("s_wait_asynccnt %0" :: "n"(0) : "memory");
            }
            __syncthreads();                // segment j landed in buf[j&1]
        }
        float p1 = -__builtin_inff();
#pragma unroll
        for (int t = 0; t < LF; ++t) {
            const int q = (j - 1) * LF + t;               // stage-1 output index
            if (q < M1) consume(q, fmaxf(S1[t], p1));
            if (has_read) {
                const float v = smem[(j & 1) * (LF * TC) + t * TC + tid];
                p1 = fmaxf(p1, v);
                B1[t] = v;
            }
        }
        if (has_read) {
            __syncthreads();                // all waves done reading buf[j&1]
            if (j + 2 < NSEG) issue_seg(j + 2);
            S1[LF - 1] = B1[LF - 1];
#pragma unroll
            for (int t = LF - 2; t >= 0; --t) S1[t] = fmaxf(B1[t], S1[t + 1]);
        }
    }
    consume(M1, __builtin_inff());          // flush final output (q2 = CH-1)
}

extern "C" void kernel_launch(void* const* d_in, const int* in_sizes, int n_in,
                              void* d_out, int out_size, void* d_ws, size_t ws_size,
                              hipStream_t stream) {
    (void)in_sizes; (void)n_in; (void)out_size; (void)d_ws; (void)ws_size;
    const float* x = (const float*)d_in[0];
    float* out     = (float*)d_out;
    dim3 grid(NCOL / TC, (NROW + CH - 1) / CH, NBATCH);
    dim3 block(TC);
    hipLaunchKernelGGL(moving_maxmin_kernel, grid, block, 0, stream, x, out);
}